// VectorQuantizerEMA_65352222376130
// MI455X (gfx1250) — compile-verified
//
#include <hip/hip_runtime.h>
#include <hip/hip_bf16.h>
#include <math.h>

typedef __attribute__((ext_vector_type(2))) float v2f;
typedef __attribute__((ext_vector_type(8))) float v8f;

#define N_PIX   65536            // 16 * 64 * 64 pixels
#define CDIM    256              // embedding dim
#define KCODES  1024             // codebook size
#define HW      4096             // 64*64
#define Q_ELEMS 16777216         // 16*256*64*64

// workspace layout (4-byte units)
#define WS_IDX    0              // int[65536]   encoding indices
#define WS_ENORM  65536          // float[1024]  ||e_k||^2
#define WS_COUNT  66560          // int[1024]    histogram
#define WS_LOSS   67584          // float[1]     loss accumulator

// ---------------------------------------------------------------------------
// init: zero histogram + loss accumulator (harness does not re-poison ws)
// ---------------------------------------------------------------------------
__global__ void vq_init(int* __restrict__ counts, float* __restrict__ lossAcc) {
  int t = threadIdx.x;
  if (t < KCODES) counts[t] = 0;
  if (t == 0) *lossAcc = 0.f;
}

// ---------------------------------------------------------------------------
// ||e_k||^2 for each codebook row (1 MB read, trivial)
// ---------------------------------------------------------------------------
__global__ void vq_enorm(const float* __restrict__ E, float* __restrict__ eNorm) {
  int k = blockIdx.x * blockDim.x + threadIdx.x;
  if (k >= KCODES) return;
  const float* row = E + (size_t)k * CDIM;
  float s = 0.f;
  #pragma unroll 8
  for (int d = 0; d < CDIM; ++d) { float v = row[d]; s += v * v; }
  eNorm[k] = s;
}

// ---------------------------------------------------------------------------
// argmin kernel: fp32 WMMA GEMM (X[65536x256] * E^T[256x1024]) fused with
// per-row running argmin of (||e||^2 - 2 x.e).
//
// One wave owns a 16-row x 128-code tile: 8 live 16x16 f32 accumulators.
// Per d-step (K+=4): 1 A fragment (2 scalar loads, reused by 8 WMMAs) and
// 8 B fragments (b64 loads, immediate-offset addressed) feed 8 independent
// v_wmma_f32_16x16x4_f32 ops (no D->A/B RAW hazards between neighbors).
// X strip is re-read only KCODES/128 = 8 times, all from L2 (X = 67MB < 192MB).
//
// V_WMMA_F32_16X16X4_F32 fragment layouts (ISA 7.12.2):
//   A 16x4 : lanes 0-15 M=lane  K={0,1} in V0,V1 ; lanes 16-31 K={2,3}
//   B 4x16 : lanes 0-15 N=lane  rows K={0,1}     ; lanes 16-31 rows K={2,3}
//   C 16x16: VGPR r, lanes 0-15 -> M=r, N=lane ; lanes 16-31 -> M=r+8
// ---------------------------------------------------------------------------
__global__ void __launch_bounds__(256) vq_argmin(
    const float* __restrict__ X, const float* __restrict__ E,
    const float* __restrict__ eNorm, int* __restrict__ idxOut) {
  const int lane = threadIdx.x & 31;
  const int wave = threadIdx.x >> 5;
  const int rsel = lane & 15;          // A: row in strip;  B: code in chunk
  const int hi   = lane >> 4;          // K-half select (adds +2 to K offset)
  const int row0 = blockIdx.x * 128 + wave * 16;

  const int n = row0 + rsel;           // pixel id handled by this lane (A side)
  const int b = n >> 12;               // n / 4096
  const int p = n & (HW - 1);
  const float* xbase = X + (size_t)b * CDIM * HW + p;   // + c*HW for channel c

  float minV[8];
  int   minI[8];
  #pragma unroll
  for (int r = 0; r < 8; ++r) { minV[r] = 3.4e38f; minI[r] = 0; }

  for (int kc = 0; kc < KCODES; kc += 128) {            // 8 outer iterations
    v8f acc[8];
    #pragma unroll
    for (int c = 0; c < 8; ++c) acc[c] = (v8f){};
    const float* ek = E + (size_t)(kc + rsel) * CDIM;   // chunk c: + c*16*CDIM

    #pragma unroll 2
    for (int d = 0; d < CDIM; d += 4) {
      const int da = d + 2 * hi;
      v2f a;                                            // one A frag, 8 uses
      a.x = xbase[(size_t)da * HW];                     // coalesced over p
      a.y = xbase[(size_t)(da + 1) * HW];
      #pragma unroll
      for (int c = 0; c < 8; ++c) {
        v2f bb = *(const v2f*)(ek + c * 16 * CDIM + da);  // imm-offset b64
        acc[c] = __builtin_amdgcn_wmma_f32_16x16x4_f32(
            /*neg_a=*/false, a, /*neg_b=*/false, bb,
            /*c_mod=*/(short)0, acc[c], /*reuse_a=*/false, /*reuse_b=*/false);
      }
    }

    #pragma unroll
    for (int c = 0; c < 8; ++c) {                       // ascending code order
      const int  code = kc + c * 16 + rsel;             // N = lane&15 matches C
      const float en  = eNorm[code];
      #pragma unroll
      for (int r = 0; r < 8; ++r) {
        float dist = en - 2.f * acc[c][r];
        if (dist < minV[r]) { minV[r] = dist; minI[r] = code; }
      }
    }
  }

  // min-reduce over the 16 lanes of each half (xor 8,4,2,1 stays in-half);
  // tie-break on smaller code index to match jnp.argmin (first occurrence).
  #pragma unroll
  for (int m = 8; m >= 1; m >>= 1) {
    #pragma unroll
    for (int r = 0; r < 8; ++r) {
      float ov = __shfl_xor(minV[r], m, 32);
      int   oi = __shfl_xor(minI[r], m, 32);
      if (ov < minV[r] || (ov == minV[r] && oi < minI[r])) {
        minV[r] = ov; minI[r] = oi;
      }
    }
  }
  if (rsel == 0) {                     // lane 0 -> rows r ; lane 16 -> rows r+8
    #pragma unroll
    for (int r = 0; r < 8; ++r) idxOut[row0 + hi * 8 + r] = minI[r];
  }
}

// ---------------------------------------------------------------------------
// gather codebook rows, transpose to (b,c,h,w), straight-through value
// out = x + (q - x)  (bit-exact vs reference), block-reduced commitment loss
// ---------------------------------------------------------------------------
__global__ void __launch_bounds__(256) vq_quant_loss(
    const float* __restrict__ X, const float* __restrict__ E,
    const int* __restrict__ idx, float* __restrict__ qOut,
    float* __restrict__ lossAcc) {
  const int lin = blockIdx.x * blockDim.x + threadIdx.x;  // (b,c,h,w) linear
  const int p = lin & (HW - 1);
  const int c = (lin >> 12) & (CDIM - 1);
  const int b = lin >> 20;
  const int n = (b << 12) | p;
  const float x = X[lin];
  const float q = E[(size_t)idx[n] * CDIM + c];           // gather, L2-resident
  const float d = q - x;
  qOut[lin] = x + d;                                      // straight-through
  __shared__ float red[256];
  red[threadIdx.x] = d * d;
  __syncthreads();
  for (int s = 128; s > 0; s >>= 1) {
    if (threadIdx.x < s) red[threadIdx.x] += red[threadIdx.x + s];
    __syncthreads();
  }
  if (threadIdx.x == 0) atomicAdd(lossAcc, red[0]);
}

// ---------------------------------------------------------------------------
// histogram of code usage (LDS-privatized, then merged)
// ---------------------------------------------------------------------------
__global__ void __launch_bounds__(256) vq_hist(
    const int* __restrict__ idx, int* __restrict__ counts) {
  __shared__ int h[KCODES];
  for (int i = threadIdx.x; i < KCODES; i += blockDim.x) h[i] = 0;
  __syncthreads();
  const int t0 = blockIdx.x * blockDim.x * 4 + threadIdx.x;
  #pragma unroll
  for (int j = 0; j < 4; ++j) {
    int t = t0 + j * blockDim.x;
    if (t < N_PIX) atomicAdd(&h[idx[t]], 1);
  }
  __syncthreads();
  for (int i = threadIdx.x; i < KCODES; i += blockDim.x)
    if (h[i]) atomicAdd(&counts[i], h[i]);
}

// ---------------------------------------------------------------------------
// one-hot encodings (268 MB write — the dominant HBM cost); 8B vector stores
// (encodings region base is 8-byte aligned: float offset 16777218)
// ---------------------------------------------------------------------------
__global__ void __launch_bounds__(256) vq_onehot(
    const int* __restrict__ idx, float* __restrict__ enc) {
  const int gid = blockIdx.x * blockDim.x + threadIdx.x;  // N_PIX * 64 threads
  const int nn = gid >> 6;
  const int k0 = (gid & 63) << 4;                         // 16 codes per thread
  const int my = idx[nn];
  float* dst = enc + (size_t)nn * KCODES + k0;
  #pragma unroll
  for (int j = 0; j < 16; j += 2) {
    v2f v;
    v.x = (k0 + j     == my) ? 1.f : 0.f;
    v.y = (k0 + j + 1 == my) ? 1.f : 0.f;
    *(v2f*)(dst + j) = v;
  }
}

// ---------------------------------------------------------------------------
// finalize: loss scalar and perplexity = exp(-sum p log(p+1e-10))
// ---------------------------------------------------------------------------
__global__ void __launch_bounds__(1024) vq_finalize(
    const int* __restrict__ counts, const float* __restrict__ lossAcc,
    float* __restrict__ outLoss, float* __restrict__ outPerp) {
  __shared__ float red[1024];
  const int t = threadIdx.x;
  const float pr = (float)counts[t] * (1.f / (float)N_PIX);
  red[t] = pr * logf(pr + 1e-10f);
  __syncthreads();
  for (int s = 512; s > 0; s >>= 1) {
    if (t < s) red[t] += red[t + s];
    __syncthreads();
  }
  if (t == 0) {
    *outPerp = expf(-red[0]);
    *outLoss = 0.25f * (*lossAcc) * (1.f / (float)Q_ELEMS);
  }
}

// ---------------------------------------------------------------------------
extern "C" void kernel_launch(void* const* d_in, const int* in_sizes, int n_in,
                              void* d_out, int out_size, void* d_ws, size_t ws_size,
                              hipStream_t stream) {
  (void)in_sizes; (void)n_in; (void)out_size; (void)ws_size;
  const float* X = (const float*)d_in[0];   // inputs (16,256,64,64) fp32
  const float* E = (const float*)d_in[1];   // embedding (1024,256) fp32

  float* out  = (float*)d_out;
  float* loss = out;                        // [0]
  float* q    = out + 1;                    // [1 .. 16777216]
  float* perp = out + 1 + Q_ELEMS;          // [16777217]
  float* enc  = out + 2 + Q_ELEMS;          // [16777218 ..] (8B aligned)

  int*   idx     = (int*)d_ws + WS_IDX;
  float* eNorm   = (float*)d_ws + WS_ENORM;
  int*   counts  = (int*)d_ws + WS_COUNT;
  float* lossAcc = (float*)d_ws + WS_LOSS;

  vq_init      <<<1, 1024, 0, stream>>>(counts, lossAcc);
  vq_enorm     <<<KCODES / 256, 256, 0, stream>>>(E, eNorm);
  vq_argmin    <<<N_PIX / 128, 256, 0, stream>>>(X, E, eNorm, idx);
  vq_quant_loss<<<Q_ELEMS / 256, 256, 0, stream>>>(X, E, idx, q, lossAcc);
  vq_hist      <<<N_PIX / (256 * 4), 256, 0, stream>>>(idx, counts);
  vq_onehot    <<<(N_PIX * 64) / 256, 256, 0, stream>>>(idx, enc);
  vq_finalize  <<<1, 1024, 0, stream>>>(counts, lossAcc, loss, perp);
}